// GaussianScaleReadout_73186242724527
// MI455X (gfx1250) — compile-verified
//
#include <hip/hip_runtime.h>

typedef float v2f __attribute__((ext_vector_type(2)));
typedef float v8f __attribute__((ext_vector_type(8)));

#define C_H      128
#define JTOT     1024            // C_H * K  (flattened reduction dim, j = c*8 + k)
#define NCHUNK   256             // JTOT / 4 (K=4 per f32 WMMA)
#define NPAIR    128             // chunk pairs (even+odd parity)
#define NB_CHUNK (NCHUNK + 2)    // +2 bias fold-in chunks
#define LOG_2_OVER_INP (-4.852030263919617f)   // log(2/256)
#define INV_LOG_SMAX   (0.2940126392944462f)   // 1/log(30)
#define NEG_INV_2SIG2  (-0.78125f)             // -1/(2*0.8^2)

// B table layout: [chunk][pair(hi)][ncol 0..3][kk 0..1]; ncol 3 = zero pad.
__device__ __forceinline__ int bidx(int chunk, int p, int nc) {
    return ((chunk * 2 + p) * 4 + nc) * 2;
}

__global__ __launch_bounds__(128)
void gsr_wmma_kernel(const float* __restrict__ h,
                     const float* __restrict__ cell,
                     const float* __restrict__ W,      // [K,3,C_H]
                     const float* __restrict__ bias,   // [K,3]
                     float* __restrict__ out,          // [BQ,3]
                     int BQ)
{
    __shared__ float Blds[NB_CHUNK * 2 * 4 * 2];   // 16512 B

    // ---- build padded B table in LDS ----
    for (int e = threadIdx.x; e < NB_CHUNK * 16; e += blockDim.x) {
        int chunk = e >> 4;
        int r     = e & 15;
        int p     = r >> 3;
        int nc    = (r >> 1) & 3;
        int kk    = r & 1;
        int kl    = p * 2 + kk;
        float v = 0.0f;
        if (nc < 3) {
            if (chunk < NCHUNK) {
                int j = chunk * 4 + kl;          // j = c*8 + k
                int c = j >> 3;
                int k = j & 7;
                v = W[k * (3 * C_H) + nc * C_H + c];
            } else {
                int k = (chunk - NCHUNK) * 4 + kl;
                v = bias[k * 3 + nc];
            }
        }
        Blds[((chunk * 2 + p) * 4 + nc) * 2 + kk] = v;
    }
    __syncthreads();

    const int lane   = threadIdx.x & 31;
    const int wave   = threadIdx.x >> 5;
    const int n      = lane & 15;
    const int hi     = lane >> 4;
    const int klocal = hi * 2;
    const int nc     = n < 3 ? n : 3;       // clamp into padded (zero) column

    const int tileBase = (blockIdx.x * 4 + wave) * 4;   // 4 tiles of 16 queries / wave

    // ---- per-tile alphas (Gaussian softmax over scales) ----
    float aE0[4], aE1[4], aO0[4], aO1[4];
    const float* pA[4];
#pragma unroll
    for (int t = 0; t < 4; ++t) {
        int q  = (tileBase + t) * 16 + n;   // A-matrix row M == lane&15
        int qc = q < BQ ? q : (BQ - 1);
        float c0 = cell[(size_t)qc * 2 + 0];
        float c1 = cell[(size_t)qc * 2 + 1];
        float cg = fmaxf(__builtin_sqrtf(c0 * c1), 1e-10f);
        float log_s = LOG_2_OVER_INP - __logf(cg);
        float tau   = 7.0f * log_s * INV_LOG_SMAX;
        tau = fminf(fmaxf(tau, 0.0f), 7.0f);

        float ex[8];
        float s = 0.0f;
#pragma unroll
        for (int k = 0; k < 8; ++k) {
            float d = (float)k - tau;
            ex[k] = __expf(d * d * NEG_INV_2SIG2);
            s += ex[k];
        }
        float inv = 1.0f / s;
        aE0[t] = (hi ? ex[2] : ex[0]) * inv;
        aE1[t] = (hi ? ex[3] : ex[1]) * inv;
        aO0[t] = (hi ? ex[6] : ex[4]) * inv;
        aO1[t] = (hi ? ex[7] : ex[5]) * inv;

        pA[t] = h + (size_t)qc * JTOT + klocal;
    }

    v8f acc[4] = {};

    // ---- software-pipelined main stream: ping-pong chunk-pair buffers ----
    auto loadPair = [&](int pr, v2f Ae[4], v2f Ao[4], v2f& Be, v2f& Bo) {
        int ch = pr * 2;
        Be = *(const v2f*)&Blds[bidx(ch,     hi, nc)];
        Bo = *(const v2f*)&Blds[bidx(ch + 1, hi, nc)];
#pragma unroll
        for (int t = 0; t < 4; ++t) {
            Ae[t] = __builtin_nontemporal_load((const v2f*)(pA[t] + (size_t)ch * 4));
            Ao[t] = __builtin_nontemporal_load((const v2f*)(pA[t] + (size_t)ch * 4 + 4));
        }
    };
    auto consumePair = [&](const v2f Ae[4], const v2f Ao[4], v2f Be, v2f Bo) {
#pragma unroll
        for (int t = 0; t < 4; ++t) {
            v2f A = Ae[t];
            A.x *= aE0[t];  A.y *= aE1[t];
            acc[t] = __builtin_amdgcn_wmma_f32_16x16x4_f32(
                false, A, false, Be, (short)0, acc[t], false, false);
        }
#pragma unroll
        for (int t = 0; t < 4; ++t) {
            v2f A = Ao[t];
            A.x *= aO0[t];  A.y *= aO1[t];
            acc[t] = __builtin_amdgcn_wmma_f32_16x16x4_f32(
                false, A, false, Bo, (short)0, acc[t], false, false);
        }
    };

    v2f A0e[4], A0o[4], A1e[4], A1o[4];
    v2f B0e, B0o, B1e, B1o;

    loadPair(0, A0e, A0o, B0e, B0o);                 // prologue
    for (int p = 0; p < NPAIR; p += 2) {
        loadPair(p + 1, A1e, A1o, B1e, B1o);         // prefetch odd slot
        consumePair(A0e, A0o, B0e, B0o);             // consume even slot
        int pn = (p + 2 < NPAIR) ? (p + 2) : (NPAIR - 1);   // clamped (no OOB)
        loadPair(pn, A0e, A0o, B0e, B0o);            // prefetch even slot
        consumePair(A1e, A1o, B1e, B1o);             // consume odd slot
    }

    // ---- bias fold-in: two extra K-chunks where A == alpha, B == bias ----
    {
        v2f B = *(const v2f*)&Blds[bidx(NCHUNK, hi, nc)];
#pragma unroll
        for (int t = 0; t < 4; ++t) {
            v2f A; A.x = aE0[t]; A.y = aE1[t];
            acc[t] = __builtin_amdgcn_wmma_f32_16x16x4_f32(
                false, A, false, B, (short)0, acc[t], false, false);
        }
        B = *(const v2f*)&Blds[bidx(NCHUNK + 1, hi, nc)];
#pragma unroll
        for (int t = 0; t < 4; ++t) {
            v2f A; A.x = aO0[t]; A.y = aO1[t];
            acc[t] = __builtin_amdgcn_wmma_f32_16x16x4_f32(
                false, A, false, B, (short)0, acc[t], false, false);
        }
    }

    // ---- writeback: C layout VGPR r -> (M = r + 8*hi, N = lane&15) ----
    if (n < 3) {
#pragma unroll
        for (int t = 0; t < 4; ++t) {
            int tile = tileBase + t;
#pragma unroll
            for (int r = 0; r < 8; ++r) {
                int b = tile * 16 + r + hi * 8;
                if (b < BQ) out[(size_t)b * 3 + n] = acc[t][r];
            }
        }
    }
}

extern "C" void kernel_launch(void* const* d_in, const int* in_sizes, int n_in,
                              void* d_out, int out_size, void* d_ws, size_t ws_size,
                              hipStream_t stream) {
    const float* h    = (const float*)d_in[0];   // [BQ, 128, 8]
    const float* cell = (const float*)d_in[1];   // [BQ, 2]
    const float* W    = (const float*)d_in[2];   // [8, 3, 128]
    const float* bias = (const float*)d_in[3];   // [8, 3]
    float* out = (float*)d_out;                  // [BQ, 3]

    int BQ = in_sizes[1] / 2;
    int queriesPerBlock = 256;                   // 4 waves * 4 tiles * 16 rows
    int blocks = (BQ + queriesPerBlock - 1) / queriesPerBlock;

    gsr_wmma_kernel<<<blocks, 128, 0, stream>>>(h, cell, W, bias, out, BQ);
}